// FactorizedSpectralConv2d_29403346108859
// MI455X (gfx1250) — compile-verified
//
#include <hip/hip_runtime.h>
#include <math.h>

// FNO factorized spectral conv via pruned-DFT-as-matmul, all f32 WMMA.
// B=16, C=64, H=W=256, modes 64(kx)x32(ky), rank 16.
// Twiddle identities: sin(t)=ctab[(j+192)&255], -sin(t)=ctab[(j+64)&255]
// where ctab[j]=cos(2*pi*j/256). All inner loops branch-free, indices
// strength-reduced to incremental adds.

typedef __attribute__((ext_vector_type(2))) float v2f;
typedef __attribute__((ext_vector_type(8))) float v8f;

#define BATCH 16
#define CHN   64
#define HH    256
#define WW    256
#define MODE1 32
#define MODE2 32
#define NMODE 64   // 2*MODE1 kx modes (0..31 and 224..255)
#define RK    16

// ---------------- WMMA helpers (V_WMMA_F32_16X16X4_F32) ----------------
// A: 16x4 f32 (2 VGPR). lanes 0-15: M=lane, K={0,1}; lanes 16-31: M=lane-16, K={2,3}
// B: 4x16 f32 (2 VGPR). VGPR0: K = 2*(lane>>4); VGPR1: K = 2*(lane>>4)+1; N = lane&15
// C/D: 16x16 f32 (8 VGPR). VGPR j: M = j + 8*(lane>>4); N = lane&15

__device__ __forceinline__ v8f wmma4(v2f a, v2f b, v8f c) {
  return __builtin_amdgcn_wmma_f32_16x16x4_f32(false, a, false, b, (short)0, c,
                                               false, false);
}

__device__ __forceinline__ v2f ldA(const float* S, int ld, int mb, int kb, int lane) {
  int m = mb + (lane & 15);
  int k = kb + ((lane >> 4) << 1);
  v2f a = { S[m * ld + k], S[m * ld + k + 1] };
  return a;
}

__device__ __forceinline__ v2f ldB(const float* S, int ld, int kb, int nb, int lane) {
  int n = nb + (lane & 15);
  int k = kb + ((lane >> 4) << 1);
  v2f b = { S[k * ld + n], S[(k + 1) * ld + n] };
  return b;
}

__device__ __forceinline__ void stD(float* D, int ld, int mb, int nb, int lane, v8f d) {
  int n = nb + (lane & 15);
  int m = mb + ((lane >> 4) << 3);
#pragma unroll
  for (int j = 0; j < 8; ++j) D[(size_t)(m + j) * ld + n] = d[j];
}

#define V8ZERO {0.f,0.f,0.f,0.f,0.f,0.f,0.f,0.f}

// ---------------- Kernel 1: fused pruned rfft2 per (b,c) image ----------------
__global__ __launch_bounds__(256) void k_fwd(const float* __restrict__ x,
                                             float* __restrict__ Xr,
                                             float* __restrict__ Xi) {
  __shared__ float ctab[256];
  __shared__ float img[64 * 260];   // 64-row chunk, stride 260 (==4 mod 64, A-use)
  __shared__ float s1[256 * 80];    // [h][64]: cols 0..31 re, 32..63 im (B-use)
  const int tid = threadIdx.x, lane = tid & 31, wv = tid >> 5;
  const int bc = blockIdx.x;
  const int g2 = (lane >> 4) << 1;

  ctab[tid] = cosf((float)tid * (6.28318530717958647692f / 256.0f));

  // Stage A: S1[h, n] = sum_w x[h,w] * B[w,n]; n<32: cos, n>=32: -sin (phase +64)
  for (int chunk = 0; chunk < 4; ++chunk) {
    __syncthreads();
    const float* src = x + ((size_t)bc * HH + chunk * 64) * WW;
    for (int i = tid; i < 64 * 64; i += 256) {
      int r = i >> 6, c4 = (i & 63) << 2;
      float4 v = *(const float4*)(src + r * WW + c4);
      *(float4*)(&img[r * 260 + c4]) = v;
    }
    __syncthreads();
    for (int t = wv; t < 16; t += 8) {
      int htl = t >> 2, ntl = t & 3;
      int n = ntl * 16 + (lane & 15);
      int f = n & 31;
      int p = (n < 32) ? 0 : 64;           // cos vs -sin phase offset
      int idx = (g2 * f + p) & 255;        // (k*f+p) at kb=0
      int stp = (f << 2) & 255;            // +4f per K step
      v8f acc = V8ZERO;
#pragma unroll 4
      for (int kb = 0; kb < 256; kb += 4) {
        v2f a = ldA(img, 260, htl * 16, kb, lane);
        v2f b = { ctab[idx], ctab[(idx + f) & 255] };
        acc = wmma4(a, b, acc);
        idx = (idx + stp) & 255;
      }
      stD(s1 + (chunk * 64) * 80, 80, htl * 16, ntl * 16, lane, acc);
    }
  }
  __syncthreads();

  // Stage B: X[m,ky] = sum_h S1[h,*] e^{-2pi i h f(m)/256}, f(m)=m<32?m:m+192
  {
    float* xrb = Xr + (size_t)bc * NMODE * MODE2;
    float* xib = Xi + (size_t)bc * NMODE * MODE2;
    int mt = wv >> 1, nt = wv & 1;
    int mm = mt * 16 + (lane & 15);
    int fm = (mm < 32) ? mm : (mm + 192);
    int idx = (g2 * fm) & 255;
    int stp = (fm << 2) & 255;
    v8f Pa = V8ZERO, Qa = V8ZERO, Ra = V8ZERO, Sa = V8ZERO;
#pragma unroll 4
    for (int kb = 0; kb < 256; kb += 4) {
      int i1 = (idx + fm) & 255;
      v2f ac = { ctab[idx], ctab[i1] };
      v2f as = { ctab[(idx + 192) & 255], ctab[(i1 + 192) & 255] };
      v2f br = ldB(s1, 80, kb, nt * 16, lane);
      v2f bi = ldB(s1, 80, kb, nt * 16 + 32, lane);
      Pa = wmma4(ac, br, Pa);   // cos * Re
      Sa = wmma4(as, bi, Sa);   // sin * Im
      Qa = wmma4(ac, bi, Qa);   // cos * Im
      Ra = wmma4(as, br, Ra);   // sin * Re
      idx = (idx + stp) & 255;
    }
    v8f re = Pa + Sa;   // Re X = sum Sr cos + Si sin
    v8f im = Qa - Ra;   // Im X = sum Si cos - Sr sin
    stD(xrb, MODE2, mt * 16, nt * 16, lane, re);
    stD(xib, MODE2, mt * 16, nt * 16, lane, im);
  }
}

// ---------------- Kernel 2: t[b,r,m,ky] = sum_i X[b,i,m,ky] * w_1[i,r,m] ----------------
__global__ __launch_bounds__(256) void k_mix1(const float* __restrict__ Xr,
                                              const float* __restrict__ Xi,
                                              const float2* __restrict__ w11,
                                              const float2* __restrict__ w21,
                                              float* __restrict__ Tr,
                                              float* __restrict__ Ti) {
  __shared__ float Ar[128 * 68], Ai[128 * 68];   // stride 68 (==4 mod 64, A-use)
  __shared__ float Wr[64 * 24], Wi[64 * 24];     // stride 24 (B-use)
  const int tid = threadIdx.x, lane = tid & 31, wv = tid >> 5;
  const int kx = blockIdx.x;
  const int bq = blockIdx.y;
  const float2* wsrc = (kx < MODE1) ? w11 : w21;
  const int kxw = kx & 31;

  for (int i = tid; i < 64 * 16; i += 256) {
    int c = i >> 4, r = i & 15;
    float2 z = wsrc[(c * RK + r) * MODE1 + kxw];
    Wr[c * 24 + r] = z.x;
    Wi[c * 24 + r] = z.y;
  }
  for (int i = tid; i < 4 * 32 * 64; i += 256) {
    int lb = i >> 11;
    int c  = (i >> 5) & 63;
    int ky = i & 31;
    int b  = bq * 4 + lb;
    size_t g = ((size_t)(b * CHN + c) * NMODE + kx) * MODE2 + ky;
    Ar[(lb * 32 + ky) * 68 + c] = Xr[g];
    Ai[(lb * 32 + ky) * 68 + c] = Xi[g];
  }
  __syncthreads();

  for (int t = wv; t < 8; t += 8) {
    v8f Pa = V8ZERO, Qa = V8ZERO, Ra = V8ZERO, Sa = V8ZERO;
#pragma unroll 4
    for (int kb = 0; kb < 64; kb += 4) {
      v2f ar = ldA(Ar, 68, t * 16, kb, lane);
      v2f ai = ldA(Ai, 68, t * 16, kb, lane);
      v2f br = ldB(Wr, 24, kb, 0, lane);
      v2f bi = ldB(Wi, 24, kb, 0, lane);
      Pa = wmma4(ar, br, Pa);
      Qa = wmma4(ai, bi, Qa);
      Ra = wmma4(ar, bi, Ra);
      Sa = wmma4(ai, br, Sa);
    }
    v8f tr = Pa - Qa, ti = Ra + Sa;
    int lb = t >> 1, ky0 = (t & 1) * 16;
    int b = bq * 4 + lb;
    float* trb = Tr + ((size_t)(b * NMODE + kx) * MODE2 + ky0) * RK;
    float* tib = Ti + ((size_t)(b * NMODE + kx) * MODE2 + ky0) * RK;
    stD(trb, RK, 0, 0, lane, tr);
    stD(tib, RK, 0, 0, lane, ti);
  }
}

// ---------------- Kernel 3: O[b,o,m,ky] = sum_r t[b,r,m,ky] * w_2[r,o,ky] ----------------
__global__ __launch_bounds__(256) void k_mix2(const float* __restrict__ Tr,
                                              const float* __restrict__ Ti,
                                              const float2* __restrict__ w12,
                                              const float2* __restrict__ w22,
                                              float* __restrict__ Or_,
                                              float* __restrict__ Oi_) {
  __shared__ float Ar[64 * 20], Ai[64 * 20];                      // A-use
  __shared__ float W1r[16 * 72], W1i[16 * 72];                    // B-use
  __shared__ float W2r[16 * 72], W2i[16 * 72];
  const int tid = threadIdx.x, lane = tid & 31, wv = tid >> 5;
  const int ky = blockIdx.x;
  const int b  = blockIdx.y;

  for (int i = tid; i < 16 * 64; i += 256) {
    int r = i >> 6, o = i & 63;
    float2 z1 = w12[(r * CHN + o) * MODE2 + ky];
    float2 z2 = w22[(r * CHN + o) * MODE2 + ky];
    W1r[r * 72 + o] = z1.x; W1i[r * 72 + o] = z1.y;
    W2r[r * 72 + o] = z2.x; W2i[r * 72 + o] = z2.y;
  }
  for (int i = tid; i < 64 * 16; i += 256) {
    int m = i >> 4, r = i & 15;
    size_t g = ((size_t)(b * NMODE + m) * MODE2 + ky) * RK + r;
    Ar[m * 20 + r] = Tr[g];
    Ai[m * 20 + r] = Ti[g];
  }
  __syncthreads();

  for (int t = wv; t < 16; t += 8) {
    int half = t >> 3, mt = (t >> 2) & 1, nt = t & 3;
    const float* br_ = half ? W2r : W1r;
    const float* bi_ = half ? W2i : W1i;
    int mb = half * 32 + mt * 16;
    v8f Pa = V8ZERO, Qa = V8ZERO, Ra = V8ZERO, Sa = V8ZERO;
#pragma unroll
    for (int kb = 0; kb < 16; kb += 4) {
      v2f ar = ldA(Ar, 20, mb, kb, lane);
      v2f ai = ldA(Ai, 20, mb, kb, lane);
      v2f br = ldB(br_, 72, kb, nt * 16, lane);
      v2f bi = ldB(bi_, 72, kb, nt * 16, lane);
      Pa = wmma4(ar, br, Pa);
      Qa = wmma4(ai, bi, Qa);
      Ra = wmma4(ar, bi, Ra);
      Sa = wmma4(ai, br, Sa);
    }
    v8f orr = Pa - Qa, oii = Ra + Sa;
    int o0 = nt * 16 + (lane & 15);
    int m0 = mb + ((lane >> 4) << 3);
#pragma unroll
    for (int j = 0; j < 8; ++j) {
      size_t g = ((size_t)(b * CHN + o0) * NMODE + (m0 + j)) * MODE2 + ky;
      Or_[g] = orr[j];
      Oi_[g] = oii[j];
    }
  }
}

// ---------------- Kernel 4: fused inverse (H-DFT then pruned irfft-W) per (b,o) ---------
__global__ __launch_bounds__(256) void k_inv(const float* __restrict__ Or_,
                                             const float* __restrict__ Oi_,
                                             float* __restrict__ out) {
  __shared__ float ctab[256];
  __shared__ float OBr[64 * 40], OBi[64 * 40];  // [m][ky], B-use
  __shared__ float y1[256 * 68];                // [h][64]: 0..31 sc*Re, 32..63 sc*Im
  const int tid = threadIdx.x, lane = tid & 31, wv = tid >> 5;
  const int bo = blockIdx.x;
  const int g2 = (lane >> 4) << 1;

  ctab[tid] = cosf((float)tid * (6.28318530717958647692f / 256.0f));
  for (int i = tid; i < 64 * 32; i += 256) {
    int m = i >> 5, ky = i & 31;
    size_t g = (size_t)bo * (NMODE * MODE2) + i;
    OBr[m * 40 + ky] = Or_[g];
    OBi[m * 40 + ky] = Oi_[g];
  }
  __syncthreads();

  // Stage D: y1[h,ky] = sum_m O[m,ky] * e^{+2pi i h f(m)/256}, scaled.
  // K loop split at m=32 so the frequency f(m) stays an arithmetic sequence.
  for (int t = wv; t < 32; t += 8) {
    int ht = t >> 1, nt = t & 1;
    int h = ht * 16 + (lane & 15);
    int stp = (h << 2) & 255;
    v8f Pa = V8ZERO, Qa = V8ZERO, Ra = V8ZERO, Sa = V8ZERO;

    int idx = (h * g2) & 255;                 // half 0: f = k
#pragma unroll 4
    for (int kb = 0; kb < 32; kb += 4) {
      int i1 = (idx + h) & 255;
      v2f ac = { ctab[idx], ctab[i1] };
      v2f as = { ctab[(idx + 192) & 255], ctab[(i1 + 192) & 255] };
      v2f br = ldB(OBr, 40, kb, nt * 16, lane);
      v2f bi = ldB(OBi, 40, kb, nt * 16, lane);
      Pa = wmma4(ac, br, Pa);
      Qa = wmma4(as, bi, Qa);
      Ra = wmma4(ac, bi, Ra);
      Sa = wmma4(as, br, Sa);
      idx = (idx + stp) & 255;
    }
    idx = (h * (224 + g2)) & 255;             // half 1: f = k + 192 (k>=32)
#pragma unroll 4
    for (int kb = 32; kb < 64; kb += 4) {
      int i1 = (idx + h) & 255;
      v2f ac = { ctab[idx], ctab[i1] };
      v2f as = { ctab[(idx + 192) & 255], ctab[(i1 + 192) & 255] };
      v2f br = ldB(OBr, 40, kb, nt * 16, lane);
      v2f bi = ldB(OBi, 40, kb, nt * 16, lane);
      Pa = wmma4(ac, br, Pa);
      Qa = wmma4(as, bi, Qa);
      Ra = wmma4(ac, bi, Ra);
      Sa = wmma4(as, br, Sa);
      idx = (idx + stp) & 255;
    }
    v8f re = Pa - Qa;   // Re y1 = Or cos - Oi sin
    v8f im = Ra + Sa;   // Im y1 = Oi cos + Or sin
    int ky = nt * 16 + (lane & 15);
    float sc = (ky == 0 ? 1.f : 2.f) * (1.f / 65536.f);
    int hrow = ht * 16 + ((lane >> 4) << 3);
#pragma unroll
    for (int j = 0; j < 8; ++j) {
      y1[(hrow + j) * 68 + ky]      = re[j] * sc;
      y1[(hrow + j) * 68 + 32 + ky] = im[j] * sc;
    }
  }
  __syncthreads();

  // Stage E: y[h,w] = sum_ky y1r*cos - y1i*sin; -sin via +64 phase offset.
  float* dst = out + (size_t)bo * HH * WW;
  for (int t = wv; t < 256; t += 8) {
    int ht = t >> 4, wt = t & 15;
    int w = wt * 16 + (lane & 15);
    int stp = (w << 2) & 255;
    v8f acc = V8ZERO;

    int idx = (w * g2) & 255;                 // half 0: cos * Re cols
#pragma unroll 4
    for (int kb = 0; kb < 32; kb += 4) {
      v2f a = ldA(y1, 68, ht * 16, kb, lane);
      v2f b = { ctab[idx], ctab[(idx + w) & 255] };
      acc = wmma4(a, b, acc);
      idx = (idx + stp) & 255;
    }
    idx = (w * g2 + 64) & 255;                // half 1: -sin * Im cols
#pragma unroll 4
    for (int kb = 32; kb < 64; kb += 4) {
      v2f a = ldA(y1, 68, ht * 16, kb, lane);
      v2f b = { ctab[idx], ctab[(idx + w) & 255] };
      acc = wmma4(a, b, acc);
      idx = (idx + stp) & 255;
    }
    stD(dst, WW, ht * 16, wt * 16, lane, acc);
  }
}

// ---------------- Host launcher ----------------
extern "C" void kernel_launch(void* const* d_in, const int* in_sizes, int n_in,
                              void* d_out, int out_size, void* d_ws, size_t ws_size,
                              hipStream_t stream) {
  (void)in_sizes; (void)n_in; (void)out_size; (void)ws_size;
  const float*  x   = (const float*)d_in[0];
  const float2* w11 = (const float2*)d_in[1];
  const float2* w12 = (const float2*)d_in[2];
  const float2* w21 = (const float2*)d_in[3];
  const float2* w22 = (const float2*)d_in[4];
  float* out = (float*)d_out;

  float* ws = (float*)d_ws;
  const size_t NX = (size_t)BATCH * CHN * NMODE * MODE2;  // 2,097,152 floats
  const size_t NT = (size_t)BATCH * NMODE * MODE2 * RK;   //   524,288 floats
  float* Xr  = ws;
  float* Xi  = Xr + NX;
  float* Tr  = Xi + NX;
  float* Ti  = Tr + NT;
  float* Orp = Ti + NT;
  float* Oip = Orp + NX;

  k_fwd <<<dim3(BATCH * CHN), dim3(256), 0, stream>>>(x, Xr, Xi);
  k_mix1<<<dim3(NMODE, BATCH / 4), dim3(256), 0, stream>>>(Xr, Xi, w11, w21, Tr, Ti);
  k_mix2<<<dim3(MODE2, BATCH), dim3(256), 0, stream>>>(Tr, Ti, w12, w22, Orp, Oip);
  k_inv <<<dim3(BATCH * CHN), dim3(256), 0, stream>>>(Orp, Oip, out);
}